// MultiHeadAttentionWithKVCache_71734543778302
// MI455X (gfx1250) — compile-verified
//
#include <hip/hip_runtime.h>
#include <math.h>

// ---------------------------------------------------------------------------
// MultiHeadAttention with KV cache, fp32, flash-style single pass over K/V.
//   B=8, Q=16, H=1024, NH=16, HD=64, PAST=8192, TOT=8208
// Memory-bound: 537MB of fp32 KV @ 23.3 TB/s ~ 23us floor. All matmuls use
// V_WMMA_F32_16X16X4_F32 (exact fp32). K/V tiles are streamed into LDS with
// the Tensor Data Mover (tensor_load_to_lds, TENSORcnt), double-buffered.
// ---------------------------------------------------------------------------

#define NHDS 16
#define HDIM 64
#define HID  1024
#define NQ   16
#define PASTL 8192
#define TOTL  8208

typedef __attribute__((ext_vector_type(2))) float v2f;
typedef __attribute__((ext_vector_type(8))) float v8f;
typedef __attribute__((ext_vector_type(4))) unsigned int u32x4;
typedef __attribute__((ext_vector_type(8))) int i32x8;
typedef __attribute__((ext_vector_type(4))) int i32x4;

#if __has_builtin(__builtin_amdgcn_tensor_load_to_lds) && \
    __has_builtin(__builtin_amdgcn_s_wait_tensorcnt)
#define USE_TDM 1
#pragma message("CDNA5: TDM tensor_load_to_lds path ENABLED")
#else
#define USE_TDM 0
#pragma message("CDNA5: TDM builtin missing -> direct-global fallback")
#endif

// D = A(16x4) * B(4x16) + C, full fp32.
// A: lane=M%16, half=lane/16, vgpr r = A[M][2*half+r]
// B: lane=N%16, half=lane/16, vgpr r = B[2*half+r][N]
// C/D: vgpr g = row M=g+8*half, lane%16 = N
__device__ __forceinline__ v8f wmma4(v2f a, v2f b, v8f c) {
  return __builtin_amdgcn_wmma_f32_16x16x4_f32(false, a, false, b, (short)0, c,
                                               false, false);
}

#if USE_TDM
// TDM: DMA one 16x64 fp32 tile (4KB) global -> LDS. If padK, insert 2 dwords
// of LDS padding per 64-dword row (row stride 66 floats -> conflict-free
// column-strided ds_load_b64 of K).
__device__ __forceinline__ void tdm_load_tile(const float* gsrc,
                                              const float* lds_dst, bool padK) {
  size_t ga = (size_t)gsrc;
  unsigned lo = (unsigned)__builtin_amdgcn_readfirstlane((int)(unsigned)ga);
  unsigned hi = (unsigned)__builtin_amdgcn_readfirstlane((int)(ga >> 32));
  unsigned la =
      (unsigned)__builtin_amdgcn_readfirstlane((int)(unsigned)(size_t)lds_dst);
  u32x4 g0;
  g0[0] = 1u;                                  // count=1 (valid descriptor)
  g0[1] = la;                                  // lds_addr (bytes)
  g0[2] = lo;                                  // global_addr[31:0]
  g0[3] = (hi & 0x01FFFFFFu) | 0x80000000u;    // global_addr[56:32] | type=2
  // group1: data_size=4B(2), tensor_dim0=64, tensor_dim1=16, tile=64x16,
  // tensor_dim0_stride=64.  K adds pad_enable, interval=64dw, amount=2dw.
  const int w0 = padK ? 0x03520000 : 0x00020000;
  i32x8 g1 = {w0, 0x00400000, 0x00100000, 0x00400000, 16, 64, 0, 0};
  i32x4 z4 = {0, 0, 0, 0};
#if __clang_major__ >= 23
  i32x8 z8 = {0, 0, 0, 0, 0, 0, 0, 0};
  __builtin_amdgcn_tensor_load_to_lds(g0, g1, z4, z4, z8, 0);
#else
  __builtin_amdgcn_tensor_load_to_lds(g0, g1, z4, z4, 0);
#endif
}
#endif

// ---------------------------------------------------------------------------
// Kernel A: q/k/v = hidden @ W^T + b.  One wave per 16x16 tile of [128x1024].
// ---------------------------------------------------------------------------
__global__ __launch_bounds__(32) void qkv_proj_kernel(
    const float* __restrict__ X,
    const float* __restrict__ Wq, const float* __restrict__ bq,
    const float* __restrict__ Wk, const float* __restrict__ bk,
    const float* __restrict__ Wv, const float* __restrict__ bv,
    float* __restrict__ q_ws, float* __restrict__ k_ws, float* __restrict__ v_ws) {
  const int blk  = blockIdx.x;
  const int wsel = blk >> 9;
  const int rem  = blk & 511;
  const int mbase = (rem >> 6) << 4;
  const int nbase = (rem & 63) << 4;

  const float* W    = (wsel == 0) ? Wq : (wsel == 1) ? Wk : Wv;
  const float* bias = (wsel == 0) ? bq : (wsel == 1) ? bk : bv;
  float*       out  = (wsel == 0) ? q_ws : (wsel == 1) ? k_ws : v_ws;

  const int lane = threadIdx.x;
  const int n    = lane & 15;
  const int h    = lane >> 4;
  const int co   = 2 * h;

  const float* xrow = X + (size_t)(mbase + n) * HID + co;
  const float* wrow = W + (size_t)(nbase + n) * HID + co;

  const float b0 = bias[nbase + n];
  v8f c;
#pragma unroll
  for (int g = 0; g < 8; ++g) c[g] = b0;

#pragma unroll 4
  for (int k0 = 0; k0 < HID; k0 += 4) {
    v2f a  = *(const v2f*)(xrow + k0);
    v2f bb = *(const v2f*)(wrow + k0);
    c = wmma4(a, bb, c);
  }

#pragma unroll
  for (int g = 0; g < 8; ++g) {
    const int m  = mbase + g + 8 * h;
    const int o  = nbase + n;
    const int bi = m >> 4, ii = m & 15;
    const int hh = o >> 6, d = o & 63;
    out[(((size_t)(bi * NHDS + hh)) * NQ + ii) * HDIM + d] = c[g];
  }
}

// ---------------------------------------------------------------------------
// Kernel B: flash attention partials. One single-wave WG per (b,head,seg).
// TDM double-buffers 16x64 K (padded) and V tiles in LDS; WMMA consumes them.
// ---------------------------------------------------------------------------
__global__ __launch_bounds__(32) void attn_kernel(
    const float* __restrict__ past_key, const float* __restrict__ past_val,
    const float* __restrict__ padding_mask,
    const float* __restrict__ q_ws, const float* __restrict__ k_ws,
    const float* __restrict__ v_ws, int nseg,
    float* __restrict__ pacc, float* __restrict__ pm, float* __restrict__ pl) {
#if USE_TDM
  // [K buf0 | K buf1 | V buf0 | V buf1] = 1056+1056+1024+1024 floats
  __shared__ float smem[4160];
#endif
  const int bh  = blockIdx.x / nseg;   // b*NH + head
  const int seg = blockIdx.x % nseg;
  const int b   = bh >> 4;

  const int lane = threadIdx.x;
  const int q    = lane & 15;          // query this lane tracks
  const int h    = lane >> 4;
  const int co   = 2 * h;
  const int step = nseg * 16;

  const float* kpast = past_key + ((size_t)bh) * PASTL * HDIM;
  const float* vpast = past_val + ((size_t)bh) * PASTL * HDIM;
  const float* knew  = k_ws + ((size_t)bh) * NQ * HDIM;
  const float* vnew  = v_ws + ((size_t)bh) * NQ * HDIM;
  const float* pmb   = padding_mask + (size_t)b * TOTL;

  // Preload this lane's Q operand slices (reused for every key tile).
  v2f qreg[16];
  {
    const float* qrow = q_ws + (((size_t)bh) * NQ + q) * HDIM + co;
#pragma unroll
    for (int t = 0; t < 16; ++t) qreg[t] = *(const v2f*)(qrow + 4 * t);
  }

  float mrun = -INFINITY;
  float lrun = 0.0f;
  v8f acc[4];
#pragma unroll
  for (int dc = 0; dc < 4; ++dc)
#pragma unroll
    for (int g = 0; g < 8; ++g) acc[dc][g] = 0.0f;

#if USE_TDM
  {  // prologue: DMA first tile pair into buffer 0
    const int j0 = seg * 16;
    const float* kt = (j0 < PASTL) ? kpast + (size_t)j0 * HDIM
                                   : knew + (size_t)(j0 - PASTL) * HDIM;
    const float* vt = (j0 < PASTL) ? vpast + (size_t)j0 * HDIM
                                   : vnew + (size_t)(j0 - PASTL) * HDIM;
    tdm_load_tile(kt, smem, true);
    tdm_load_tile(vt, smem + 2112, false);
  }
#endif

  int cur = 0;
  for (int j0 = seg * 16; j0 < TOTL; j0 += step) {
    const int jn = j0 + step;
#if USE_TDM
    if (jn < TOTL) {  // DMA next pair into the other buffer
      const float* kt = (jn < PASTL) ? kpast + (size_t)jn * HDIM
                                     : knew + (size_t)(jn - PASTL) * HDIM;
      const float* vt = (jn < PASTL) ? vpast + (size_t)jn * HDIM
                                     : vnew + (size_t)(jn - PASTL) * HDIM;
      tdm_load_tile(kt, smem + (cur ? 0 : 1056), true);
      tdm_load_tile(vt, smem + 2112 + (cur ? 0 : 1024), false);
      __builtin_amdgcn_s_wait_tensorcnt((short)2);  // current pair done
    } else {
      __builtin_amdgcn_s_wait_tensorcnt((short)0);
    }
    const float* kt_l = smem + (cur ? 1056 : 0);           // row stride 66
    const float* vt_l = smem + 2112 + (cur ? 1024 : 0);    // row stride 64
#else
    const float *kbase, *vbase;
    if (j0 < PASTL) {
      kbase = kpast + (size_t)j0 * HDIM;
      vbase = vpast + (size_t)j0 * HDIM;
      if (jn < PASTL) {
        __builtin_prefetch(kpast + (size_t)jn * HDIM + lane * 32, 0, 1);
        __builtin_prefetch(vpast + (size_t)jn * HDIM + lane * 32, 0, 1);
      }
    } else {
      kbase = knew + (size_t)(j0 - PASTL) * HDIM;
      vbase = vnew + (size_t)(j0 - PASTL) * HDIM;
    }
    const float* krow = kbase + q * HDIM + co;
#endif

    // ---- S^T = K(16x64) * Q^T(64x16): A rows = keys, B cols = queries ----
    v8f st = {};
#pragma unroll
    for (int k0 = 0; k0 < HDIM; k0 += 4) {
#if USE_TDM
      v2f a = *(const v2f*)(kt_l + q * 66 + k0 + co);
#else
      v2f a = *(const v2f*)(krow + k0);
#endif
      st = wmma4(a, qreg[k0 >> 2], st);
    }
    // lane holds score(query=q, key=j0+g+8h) in st[g]

    // ---- scale + causal + padding masks ----
    float p8[8];
    float tmax = -INFINITY;
#pragma unroll
    for (int g = 0; g < 8; ++g) {
      const int j = j0 + g + 8 * h;
      float s = st[g] * 0.125f;                 // 1/sqrt(64)
      if (j > PASTL + q) s += -1e9f;            // causal
      s += pmb[j] * (-1e9f);                    // padding
      p8[g] = s;
      tmax = fmaxf(tmax, s);
    }
    tmax = fmaxf(tmax, __shfl_xor(tmax, 16, 32));

    // ---- online softmax update (per-lane = per-query) ----
    const float mnew = fmaxf(mrun, tmax);
    const float corr = __expf(mrun - mnew);
    float lsum = 0.0f;
#pragma unroll
    for (int g = 0; g < 8; ++g) {
      p8[g] = __expf(p8[g] - mnew);
      lsum += p8[g];
    }
    lsum += __shfl_xor(lsum, 16, 32);
    lrun = lrun * corr + lsum;
    mrun = mnew;

    // rescale accumulator rows (acc row M = g+8h)
#pragma unroll
    for (int g = 0; g < 8; ++g) {
      const float rc = __shfl(corr, g + 8 * h, 32);
#pragma unroll
      for (int dc = 0; dc < 4; ++dc) acc[dc][g] *= rc;
    }

    // ---- acc += P(16x16) * V(16x64) ----
#pragma unroll
    for (int s = 0; s < 4; ++s) {
      const int sh  = s >> 1;
      const int src = q + 16 * sh;
      const float x0 = __shfl(p8[(4 * s + 0) & 7], src, 32);
      const float x1 = __shfl(p8[(4 * s + 1) & 7], src, 32);
      const float x2 = __shfl(p8[(4 * s + 2) & 7], src, 32);
      const float x3 = __shfl(p8[(4 * s + 3) & 7], src, 32);
      v2f av;
      av.x = h ? x2 : x0;                        // P[q][4s+2h]
      av.y = h ? x3 : x1;                        // P[q][4s+2h+1]
#if USE_TDM
      const float* v0 = vt_l + (4 * s + 2 * h) * HDIM + q;
#else
      const float* v0 = vbase + (4 * s + 2 * h) * HDIM + q;
#endif
#pragma unroll
      for (int dc = 0; dc < 4; ++dc) {
        v2f bvv;
        bvv.x = v0[16 * dc];
        bvv.y = v0[16 * dc + HDIM];
        acc[dc] = wmma4(av, bvv, acc[dc]);
      }
    }
    cur ^= 1;
  }

  // ---- write this segment's flash partial ----
  float* pa = pacc + ((size_t)bh * nseg + seg) * (NQ * HDIM);
#pragma unroll
  for (int g = 0; g < 8; ++g) {
    const int row = g + 8 * h;
#pragma unroll
    for (int dc = 0; dc < 4; ++dc) pa[row * HDIM + 16 * dc + q] = acc[dc][g];
  }
  if (h == 0) {
    pm[((size_t)bh * nseg + seg) * NQ + q] = mrun;
    pl[((size_t)bh * nseg + seg) * NQ + q] = lrun;
  }
}

// ---------------------------------------------------------------------------
// Kernel B2: combine segment partials (log-sum-exp) and write the permuted
// layout the reference feeds into Wo (transpose(1,-1).reshape).
// ---------------------------------------------------------------------------
__global__ __launch_bounds__(256) void combine_kernel(
    const float* __restrict__ pacc, const float* __restrict__ pm,
    const float* __restrict__ pl, int nseg, float* __restrict__ x_ws) {
  const int bh = blockIdx.x;
  const int b  = bh >> 4;
  const int hh = bh & 15;
  for (int e = threadIdx.x; e < NQ * HDIM; e += 256) {
    const int qq = e >> 6;
    const int d  = e & 63;
    float M = -INFINITY;
    for (int s = 0; s < nseg; ++s)
      M = fmaxf(M, pm[((size_t)bh * nseg + s) * NQ + qq]);
    float num = 0.0f, den = 0.0f;
    for (int s = 0; s < nseg; ++s) {
      const float f = __expf(pm[((size_t)bh * nseg + s) * NQ + qq] - M);
      num += f * pacc[((size_t)bh * nseg + s) * (NQ * HDIM) + qq * HDIM + d];
      den += f * pl[((size_t)bh * nseg + s) * NQ + qq];
    }
    const float val = num / den;
    const int r = d >> 2;
    const int c = ((d & 3) << 8) | (qq << 4) | hh;
    x_ws[((size_t)b * 16 + r) * HID + c] = val;
  }
}

// ---------------------------------------------------------------------------
// Kernel C: out = Xperm @ Wo^T + bo  ([128 x 1024] GEMM), writes d_out.
// ---------------------------------------------------------------------------
__global__ __launch_bounds__(32) void out_proj_kernel(
    const float* __restrict__ Xw, const float* __restrict__ Wo,
    const float* __restrict__ bo, float* __restrict__ out) {
  const int blk   = blockIdx.x;
  const int mbase = (blk >> 6) << 4;
  const int nbase = (blk & 63) << 4;
  const int lane  = threadIdx.x;
  const int n     = lane & 15;
  const int h     = lane >> 4;
  const int co    = 2 * h;

  const float* xrow = Xw + (size_t)(mbase + n) * HID + co;
  const float* wrow = Wo + (size_t)(nbase + n) * HID + co;

  const float b0 = bo[nbase + n];
  v8f c;
#pragma unroll
  for (int g = 0; g < 8; ++g) c[g] = b0;

#pragma unroll 4
  for (int k0 = 0; k0 < HID; k0 += 4) {
    v2f a  = *(const v2f*)(xrow + k0);
    v2f bb = *(const v2f*)(wrow + k0);
    c = wmma4(a, bb, c);
  }

#pragma unroll
  for (int g = 0; g < 8; ++g)
    out[(size_t)(mbase + g + 8 * h) * HID + nbase + n] = c[g];
}

// ---------------------------------------------------------------------------
extern "C" void kernel_launch(void* const* d_in, const int* in_sizes, int n_in,
                              void* d_out, int out_size, void* d_ws,
                              size_t ws_size, hipStream_t stream) {
  (void)in_sizes; (void)n_in; (void)out_size;
  const float* hidden   = (const float*)d_in[0];
  const float* past_key = (const float*)d_in[1];
  const float* past_val = (const float*)d_in[2];
  // d_in[3] causal_mask applied analytically (j > PAST + i)
  const float* padding  = (const float*)d_in[4];
  const float* Wq = (const float*)d_in[5];
  const float* bq = (const float*)d_in[6];
  const float* Wk = (const float*)d_in[7];
  const float* bk = (const float*)d_in[8];
  const float* Wv = (const float*)d_in[9];
  const float* bv = (const float*)d_in[10];
  const float* Wo = (const float*)d_in[11];
  const float* bo = (const float*)d_in[12];
  float* out = (float*)d_out;

  float* ws   = (float*)d_ws;
  float* q_ws = ws;
  float* k_ws = ws + 131072;
  float* v_ws = ws + 262144;
  float* x_ws = ws + 393216;
  const size_t base = 524288;  // floats used so far

  // Pick the largest segment split whose partials fit in the workspace.
  int nseg = 1;
  const int cand[5] = {32, 16, 8, 4, 2};
  for (int i = 0; i < 5; ++i) {
    const size_t need = (base + (size_t)cand[i] * 128 * (1024 + 32)) * 4;
    if (need <= ws_size) { nseg = cand[i]; break; }
  }
  float* pacc = ws + base;                          // [128][nseg][16][64]
  float* pm   = pacc + (size_t)128 * nseg * 1024;   // [128][nseg][16]
  float* pl   = pm + (size_t)128 * nseg * 16;       // [128][nseg][16]

  qkv_proj_kernel<<<1536, 32, 0, stream>>>(hidden, Wq, bq, Wk, bk, Wv, bv,
                                           q_ws, k_ws, v_ws);
  attn_kernel<<<128 * nseg, 32, 0, stream>>>(past_key, past_val, padding, q_ws,
                                             k_ws, v_ws, nseg, pacc, pm, pl);
  combine_kernel<<<128, 256, 0, stream>>>(pacc, pm, pl, nseg, x_ws);
  out_proj_kernel<<<512, 32, 0, stream>>>(x_ws, Wo, bo, out);
}